// MemSlotAttention_29746943492649
// MI455X (gfx1250) — compile-verified
//
#include <hip/hip_runtime.h>
#include <hip/hip_bf16.h>

// MemSlotAttention for MI455X (gfx1250), fp32 path via V_WMMA_F32_16X16X4_F32.
//
// Shapes: B=16, L=4096, D_MODEL=768, N_SLOTS(K)=32, D_PROJ(P)=256, scale=1/16.
// Algebraic rewrites: S = (attn @ H) @ Wv^T  (avoids materializing V [B,L,D]);
// K-proj and rq-proj fused into one GEMM with concatenated weights.
//
// d_out layout (fp32, concatenated): H_hat [B,L,D] | attn [B,K,L] | S [B,K,D].

typedef __attribute__((ext_vector_type(2))) float v2f;
typedef __attribute__((ext_vector_type(8))) float v8f;

#define LDSTRIDE 68   // 64 + 4 pad: compute reads conflict-free (banks 4*r+c vs +2)

// ---------------------------------------------------------------------------
// Generic batched fp32 GEMM on WMMA 16x16x4.
//   BT=true : C[m,n] = sum_k X[m,k] * W[n,k]   (X [M,Kd] ldx, W [N,Kd] ldw)
//   BT=false: C[m,n] = sum_k X[m,k] * W[k,n]   (X [M,Kd] ldx, W [Kd,N] ldw)
// Requirements: M%16==0, N%32==0, Kd%32==0 (true at all call sites).
// Block = 256 threads = 8 waves; block tile = 16 (M) x 256 (N); each wave owns
// a 16x32 output strip (two 16x16 WMMA accumulators sharing the A fragment).
// Since N%32==0 and n0%32==0, (n0 < N) implies (n0+16 < N): one wave-uniform
// activity flag guards the whole strip -> branch-free inner loop, so the
// compiler can clause the B loads and stagger s_wait_loadcnt across the two
// independent WMMA accumulation chains.
// A-tiles (16x64) double-buffered in LDS: one barrier per 64-K block, next
// block's global->reg A load overlaps the 32-WMMA compute stream.
// WMMA operand layout per CDNA5 ISA 7.12.2:
//   A lane: row = lane%16, K = 2*(lane/16) + vgpr(0/1)
//   B lane: col = lane%16, K = 2*(lane/16) + vgpr(0/1)
//   D vgpr r: row = r + 8*(lane/16), col = lane%16
// ---------------------------------------------------------------------------
template <bool BT>
__global__ __launch_bounds__(256)
void gemm_f32_wmma(const float* __restrict__ X, const float* __restrict__ W,
                   float* __restrict__ C,
                   int M, int N, int Kd,
                   int ldx, int ldw, int ldc,
                   long long bsX, long long bsW, long long bsC)
{
    __shared__ float ldsA[2][16 * LDSTRIDE];

    const int tid  = threadIdx.x;
    const int lane = tid & 31;
    const int wave = tid >> 5;
    const int ln   = lane & 15;   // N index within tile (and A row for reads)
    const int hg   = lane >> 4;   // lane-half -> K pair / row+8 select
    const int m0   = blockIdx.y * 16;
    const int n0   = blockIdx.x * 256 + wave * 32;
    const bool act = (n0 < N);    // wave-uniform; covers both 16-wide subtiles

    const float* Xb = X + (long long)blockIdx.z * bsX;
    const float* Wb = W + (long long)blockIdx.z * bsW;
    float*       Cb = C + (long long)blockIdx.z * bsC;

    v8f acc0 = {0.f, 0.f, 0.f, 0.f, 0.f, 0.f, 0.f, 0.f};
    v8f acc1 = {0.f, 0.f, 0.f, 0.f, 0.f, 0.f, 0.f, 0.f};

    // Cooperative A-tile staging: 16 rows x 64 cols, one float4 per thread.
    const int arow = tid >> 4;          // 0..15
    const int ac4  = (tid & 15) * 4;    // 0,4,..,60

    // Prologue: stage block 0 into buffer 0.
    {
        float4 a0 = {0.f, 0.f, 0.f, 0.f};
        if (ac4 < Kd)
            a0 = *(const float4*)(Xb + (long long)(m0 + arow) * ldx + ac4);
        *(float4*)&ldsA[0][arow * LDSTRIDE + ac4] = a0;
    }

    const int nblk = (Kd + 63) >> 6;
    for (int ib = 0; ib < nblk; ++ib) {
        const int  kb      = ib << 6;
        const int  cur     = ib & 1;
        const bool haveNxt = (kb + 64) < Kd;

        __syncthreads();   // stage of buffer `cur` visible to all waves

        // Issue next block's global->reg A load (overlaps compute below).
        float4 anext = {0.f, 0.f, 0.f, 0.f};
        if (haveNxt && (kb + 64 + ac4 < Kd))
            anext = *(const float4*)(Xb + (long long)(m0 + arow) * ldx + kb + 64 + ac4);

        // Prefetch next B block into cache hierarchy.
        if (haveNxt) {
            if (BT) {
                __builtin_prefetch(Wb + (long long)(n0 + ln)      * ldw + kb + 64, 0, 1);
                __builtin_prefetch(Wb + (long long)(n0 + 16 + ln) * ldw + kb + 64, 0, 1);
            } else {
                __builtin_prefetch(Wb + (long long)(kb + 64 + 2 * hg) * ldw + n0 + ln, 0, 1);
            }
        }

        if (act) {   // wave-uniform: EXEC all-ones inside for WMMA
            const float* As = &ldsA[cur][ln * LDSTRIDE];
#pragma unroll
            for (int kk = 0; kk < 8; ++kk) {
                const int o = 4 * kk;
                v2f a, b0, b1;
                a.x = As[o + 2 * hg + 0];
                a.y = As[o + 2 * hg + 1];
                if (BT) {
                    const float* w0 = Wb + (long long)(n0 + ln) * ldw + (kb + o + 2 * hg);
                    b0.x = w0[0]; b0.y = w0[1];
                    const float* w1 = w0 + (long long)16 * ldw;
                    b1.x = w1[0]; b1.y = w1[1];
                } else {
                    const float* w0 = Wb + (long long)(kb + o + 2 * hg) * ldw + (n0 + ln);
                    b0.x = w0[0]; b0.y = w0[ldw];
                    b1.x = w0[16]; b1.y = w0[ldw + 16];
                }
                acc0 = __builtin_amdgcn_wmma_f32_16x16x4_f32(false, a, false, b0,
                                                             (short)0, acc0, false, false);
                acc1 = __builtin_amdgcn_wmma_f32_16x16x4_f32(false, a, false, b1,
                                                             (short)0, acc1, false, false);
            }
            if (kb + 32 < Kd) {   // second half of the 64-K block
#pragma unroll
                for (int kk = 8; kk < 16; ++kk) {
                    const int o = 4 * kk;
                    v2f a, b0, b1;
                    a.x = As[o + 2 * hg + 0];
                    a.y = As[o + 2 * hg + 1];
                    if (BT) {
                        const float* w0 = Wb + (long long)(n0 + ln) * ldw + (kb + o + 2 * hg);
                        b0.x = w0[0]; b0.y = w0[1];
                        const float* w1 = w0 + (long long)16 * ldw;
                        b1.x = w1[0]; b1.y = w1[1];
                    } else {
                        const float* w0 = Wb + (long long)(kb + o + 2 * hg) * ldw + (n0 + ln);
                        b0.x = w0[0]; b0.y = w0[ldw];
                        b1.x = w0[16]; b1.y = w0[ldw + 16];
                    }
                    acc0 = __builtin_amdgcn_wmma_f32_16x16x4_f32(false, a, false, b0,
                                                                 (short)0, acc0, false, false);
                    acc1 = __builtin_amdgcn_wmma_f32_16x16x4_f32(false, a, false, b1,
                                                                 (short)0, acc1, false, false);
                }
            }
        }

        // Stage next block into the other buffer (no reader conflict: readers
        // of buf[cur^1] only start after next iteration's barrier).
        if (haveNxt)
            *(float4*)&ldsA[cur ^ 1][arow * LDSTRIDE + ac4] = anext;
    }

    if (act) {
#pragma unroll
        for (int r = 0; r < 8; ++r) {
            Cb[(long long)(m0 + r + 8 * hg) * ldc + (n0 + ln)]      = acc0[r];
            Cb[(long long)(m0 + r + 8 * hg) * ldc + (n0 + 16 + ln)] = acc1[r];
        }
    }
}

// ---------------------------------------------------------------------------
// Masked, scaled softmax over L (in place). One block per (b,k) row.
// ---------------------------------------------------------------------------
__global__ __launch_bounds__(256)
void softmax_rows_L(float* __restrict__ logits, const unsigned char* __restrict__ mask,
                    int K, int L, float scale)
{
    const int row = blockIdx.x;          // b*K + k
    const int b   = row / K;
    float* p = logits + (long long)row * L;
    const unsigned char* mrow = mask + (long long)b * L;
    __shared__ float red[256];

    float mx = -3.4e38f;
    for (int i = threadIdx.x; i < L; i += 256) {
        float x = mrow[i] ? p[i] * scale : -1.0e9f;
        mx = fmaxf(mx, x);
    }
    red[threadIdx.x] = mx;
    __syncthreads();
    for (int s = 128; s > 0; s >>= 1) {
        if (threadIdx.x < s) red[threadIdx.x] = fmaxf(red[threadIdx.x], red[threadIdx.x + s]);
        __syncthreads();
    }
    mx = red[0];
    __syncthreads();

    float sum = 0.f;
    for (int i = threadIdx.x; i < L; i += 256) {
        float x = mrow[i] ? p[i] * scale : -1.0e9f;
        sum += __expf(x - mx);
    }
    red[threadIdx.x] = sum;
    __syncthreads();
    for (int s = 128; s > 0; s >>= 1) {
        if (threadIdx.x < s) red[threadIdx.x] += red[threadIdx.x + s];
        __syncthreads();
    }
    const float inv = 1.0f / red[0];

    for (int i = threadIdx.x; i < L; i += 256) {
        float x = mrow[i] ? p[i] * scale : -1.0e9f;
        p[i] = __expf(x - mx) * inv;
    }
}

// ---------------------------------------------------------------------------
// Softmax over K=32 (in place, scaled, no mask). One thread per token row.
// ---------------------------------------------------------------------------
__global__ __launch_bounds__(256)
void softmax_k32(float* __restrict__ RL, long long ntok, float scale)
{
    const long long t = (long long)blockIdx.x * blockDim.x + threadIdx.x;
    if (t >= ntok) return;
    float* p = RL + t * 32;
    float x[32];
#pragma unroll
    for (int i = 0; i < 8; ++i) {
        float4 q = ((const float4*)p)[i];
        x[4 * i + 0] = q.x * scale; x[4 * i + 1] = q.y * scale;
        x[4 * i + 2] = q.z * scale; x[4 * i + 3] = q.w * scale;
    }
    float mx = x[0];
#pragma unroll
    for (int i = 1; i < 32; ++i) mx = fmaxf(mx, x[i]);
    float s = 0.f;
#pragma unroll
    for (int i = 0; i < 32; ++i) { x[i] = __expf(x[i] - mx); s += x[i]; }
    const float inv = 1.0f / s;
#pragma unroll
    for (int i = 0; i < 8; ++i) {
        float4 q;
        q.x = x[4 * i + 0] * inv; q.y = x[4 * i + 1] * inv;
        q.z = x[4 * i + 2] * inv; q.w = x[4 * i + 3] * inv;
        ((float4*)p)[i] = q;
    }
}

// ---------------------------------------------------------------------------
extern "C" void kernel_launch(void* const* d_in, const int* in_sizes, int n_in,
                              void* d_out, int out_size, void* d_ws, size_t ws_size,
                              hipStream_t stream)
{
    (void)in_sizes; (void)n_in; (void)out_size; (void)ws_size;

    const float*         H     = (const float*)d_in[0];
    const unsigned char* mask  = (const unsigned char*)d_in[1];   // jnp bool -> 1 byte
    const float*         slots = (const float*)d_in[2];
    const float*         Wq    = (const float*)d_in[3];
    const float*         Wk    = (const float*)d_in[4];
    const float*         Wv    = (const float*)d_in[5];
    const float*         Wrq   = (const float*)d_in[6];
    const float*         Wrk   = (const float*)d_in[7];
    const float*         Wrv   = (const float*)d_in[8];

    const int B = 16, L = 4096, D = 768, K = 32;
    const long long BL = (long long)B * L;          // 65536
    const float scale = 0.0625f;                    // 256^-0.5

    // Output sections.
    float* out  = (float*)d_out;
    float* Hhat = out;                               // [B,L,D]
    float* attn = out + BL * D;                      // [B,K,L]
    float* Sout = attn + (long long)B * K * L;       // [B,K,D]

    // Workspace layout (floats).
    float* ws    = (float*)d_ws;
    float* KRQ   = ws;                               // [BL, 512]  (K | rq)
    float* Wcat  = KRQ   + BL * 512;                 // [512, 768] (Wk ; Wrq)
    float* Wcat2 = Wcat  + 512LL * 768;              // [1024,768] (Wrk ; Wrv)
    float* Qb    = Wcat2 + 1024LL * 768;             // [32, 256]
    float* Tb    = Qb    + 32LL * 256;               // [B*32, 768]  attn@H
    float* RKV   = Tb    + 512LL * 768;              // [B*32, 1024] (rk | rv)
    float* RL    = RKV   + 512LL * 1024;             // [B, L, 32]

    // Concatenate weights (device-to-device, graph-capture safe).
    hipMemcpyAsync(Wcat,              Wk,  sizeof(float) * 256 * 768, hipMemcpyDeviceToDevice, stream);
    hipMemcpyAsync(Wcat + 256LL*768,  Wrq, sizeof(float) * 256 * 768, hipMemcpyDeviceToDevice, stream);
    hipMemcpyAsync(Wcat2,             Wrk, sizeof(float) * 256 * 768, hipMemcpyDeviceToDevice, stream);
    hipMemcpyAsync(Wcat2 + 256LL*768, Wrv, sizeof(float) * 768 * 768, hipMemcpyDeviceToDevice, stream);

    const dim3 blk(256);

    // Q = slots @ Wq^T                  [32,256]
    gemm_f32_wmma<true><<<dim3(1, 2, 1), blk, 0, stream>>>(
        slots, Wq, Qb, 32, 256, 768, 768, 768, 256, 0, 0, 0);

    // KRQ = H @ [Wk;Wrq]^T              [BL,512]  (cols 0-255 = K, 256-511 = rq)
    gemm_f32_wmma<true><<<dim3(2, 4096, 1), blk, 0, stream>>>(
        H, Wcat, KRQ, 65536, 512, 768, 768, 768, 512, 0, 0, 0);

    // logits[b] = Q @ K_b^T -> attn section (raw, softmaxed in place next)
    gemm_f32_wmma<true><<<dim3(16, 2, 16), blk, 0, stream>>>(
        Qb, KRQ, attn, 32, 4096, 256, 256, 512, 4096,
        0, 4096LL * 512, (long long)K * L);

    // attn = softmax_L(mask ? logits*scale : -1e9)
    softmax_rows_L<<<dim3(B * K), blk, 0, stream>>>(attn, mask, K, L, scale);

    // T[b] = attn_b @ H_b               [32,768]
    gemm_f32_wmma<false><<<dim3(3, 2, 16), blk, 0, stream>>>(
        attn, H, Tb, 32, 768, 4096, 4096, 768, 768,
        (long long)K * L, (long long)L * D, 32LL * 768);

    // S = T @ Wv^T -> S output section  [B*32,768]
    gemm_f32_wmma<true><<<dim3(3, 32, 1), blk, 0, stream>>>(
        Tb, Wv, Sout, 512, 768, 768, 768, 768, 768, 0, 0, 0);

    // RKV = S @ [Wrk;Wrv]^T             [B*32,1024] (cols 0-255 = rk, 256-1023 = rv)
    gemm_f32_wmma<true><<<dim3(4, 32, 1), blk, 0, stream>>>(
        Sout, Wcat2, RKV, 512, 1024, 768, 768, 768, 1024, 0, 0, 0);

    // RL[b] = rq_b @ rk_b^T             [L,32] per batch
    gemm_f32_wmma<true><<<dim3(1, 256, 16), blk, 0, stream>>>(
        KRQ + 256, RKV, RL, 4096, 32, 256, 512, 1024, 32,
        4096LL * 512, 32LL * 1024, 4096LL * 32);

    // recon = softmax_K(RL * scale)
    softmax_k32<<<dim3(256), blk, 0, stream>>>(RL, BL, scale);

    // H_hat[b] = recon_b @ rv_b         [L,768] per batch
    gemm_f32_wmma<false><<<dim3(3, 256, 16), blk, 0, stream>>>(
        RL, RKV + 256, Hhat, 4096, 768, 32, 32, 1024, 768,
        4096LL * 32, 32LL * 1024, (long long)L * D);
}